// FarthestPointsReduce_70394513981912
// MI455X (gfx1250) — compile-verified
//
#include <hip/hip_runtime.h>
#include <stdint.h>

#ifndef __has_builtin
#define __has_builtin(x) 0
#endif

#if __has_builtin(__builtin_amdgcn_global_load_async_to_lds_b128)
#define HAVE_ASYNC_B128 1
#else
#define HAVE_ASYNC_B128 0
#endif

#if __has_builtin(__builtin_amdgcn_s_wait_asynccnt)
#define HAVE_WAIT_ASYNCCNT 1
#else
#define HAVE_WAIT_ASYNCCNT 0
#endif

#define AS1 __attribute__((address_space(1)))
#define AS3 __attribute__((address_space(3)))

// Matches the builtin's parameter pointee type (GCC vector style int4).
typedef int v4i __attribute__((vector_size(16)));

namespace {
constexpr int kN    = 32;     // batches
constexpr int kL    = 8192;   // points per cloud
constexpr int kM    = 2048;   // samples = kL * 0.25
constexpr int kD    = 128;    // feature dim
constexpr int kBlk  = 1024;   // threads per FPS block (32 waves, wave32)
constexpr int kPPT  = kL / kBlk;                        // 8 points per thread
constexpr int kChunkIters = (kL * 3 * 4) / 16 / kBlk;   // 6 x B128 per thread
constexpr int kWaves = kBlk / 32;                       // 32
}  // namespace

// Stage 16 bytes global -> LDS: ASYNCcnt-tracked DMA on gfx1250
// (GLOBAL_LOAD_ASYNC_TO_LDS_B128), bypasses VGPRs entirely.
static __device__ __forceinline__ void stage16(const float* __restrict__ g,
                                               float* l) {
#if HAVE_ASYNC_B128
  __builtin_amdgcn_global_load_async_to_lds_b128(
      (AS1 v4i*)(uintptr_t)g,   // global addr == flat addr
      (AS3 v4i*)l,              // addrspacecast generic -> LDS
      /*imm offset=*/0, /*cpol=*/0);
#else
  *reinterpret_cast<float4*>(l) = *reinterpret_cast<const float4*>(g);
#endif
}

static __device__ __forceinline__ void wait_async_lds() {
#if HAVE_ASYNC_B128
#if HAVE_WAIT_ASYNCCNT
  __builtin_amdgcn_s_wait_asynccnt(0);
#else
  asm volatile("s_wait_asynccnt 0" ::: "memory");
#endif
#endif
}

// Monotonic argmax key: squared distances are >= 0 (or +inf), so their IEEE
// bits are order-isomorphic to the values. Complemented index => on equal
// values the SMALLER index wins a u64 max (jnp.argmax first-occurrence).
static __device__ __forceinline__ unsigned long long pack_key(float v, int i) {
  return ((unsigned long long)__float_as_uint(v) << 32) |
         (unsigned)(0xFFFFFFFFu - (unsigned)i);
}
static __device__ __forceinline__ int key_index(unsigned long long k) {
  return (int)(0xFFFFFFFFu - (unsigned)(k & 0xFFFFFFFFull));
}

// ---------------------------------------------------------------------------
// Kernel 1: farthest point sampling. One 1024-thread workgroup per batch.
//  - cloud (96 KB) staged to LDS via async DMA (needed for random-access
//    winner lookup), each thread's 8 points then hoisted into registers
//  - min_d in registers; per-step body is pure VALU + one key exchange
//  - ONE barrier per step: per-wave u64-key max via shfl_xor butterfly,
//    double-buffered per-wave results in LDS, every wave redundantly
//    reduces all 32 keys itself (no wave0 funnel, no broadcast round-trip).
// ---------------------------------------------------------------------------
__global__ __launch_bounds__(kBlk, 1) void fps_kernel(
    const float* __restrict__ coords,   // [kN, kL, 3]
    float* __restrict__ coords_out,     // [kN, kM, 3]
    int* __restrict__ idx_out)          // [kN, kM] (workspace)
{
  extern __shared__ __align__(16) float smem[];
  float* pts = smem;                                        // 3*kL floats AoS
  unsigned long long* keys = (unsigned long long*)(smem + 3 * kL);  // [2][32]

  const int b = blockIdx.x;
  const int t = threadIdx.x;
  const int lane = t & 31;
  const int wid  = t >> 5;
  const float* cb = coords + (size_t)b * kL * 3;

  // --- stage the whole cloud into LDS (6144 x 16B async chunks) ---
#pragma unroll
  for (int k = 0; k < kChunkIters; ++k) {
    const int chunk = t + k * kBlk;
    stage16(cb + chunk * 4, pts + chunk * 4);
  }
  wait_async_lds();
  __syncthreads();

  // Hoist this thread's 8 points into registers (loop-invariant across all
  // 2047 steps; stride-3 word reads are bank-conflict-free: gcd(3,64)=1).
  float px[kPPT], py[kPPT], pz[kPPT], md[kPPT];
#pragma unroll
  for (int k = 0; k < kPPT; ++k) {
    const int i = t + k * kBlk;
    px[k] = pts[3 * i + 0];
    py[k] = pts[3 * i + 1];
    pz[k] = pts[3 * i + 2];
    md[k] = __builtin_inff();
  }

  // sample 0 is always point 0 (deterministic start)
  float lx = pts[0], ly = pts[1], lz = pts[2];
  if (t == 0) {
    float* co = coords_out + (size_t)b * kM * 3;
    co[0] = lx; co[1] = ly; co[2] = lz;
    idx_out[b * kM] = 0;
  }

  for (int s = 1; s < kM; ++s) {
    // fused min-distance update + thread-local argmax, pure VALU.
    // ascending k + strict > keeps the first occurrence of the max.
    float bv = -__builtin_inff();
    int   bi = 0;
#pragma unroll
    for (int k = 0; k < kPPT; ++k) {
      const float dx = px[k] - lx;
      const float dy = py[k] - ly;
      const float dz = pz[k] - lz;
      const float d  = dx * dx + dy * dy + dz * dz;
      float m = md[k];
      m = d < m ? d : m;
      md[k] = m;
      if (m > bv) { bv = m; bi = t + k * kBlk; }
    }

    // wave32 butterfly max on the packed u64 key
    unsigned long long key = pack_key(bv, bi);
#pragma unroll
    for (int off = 16; off > 0; off >>= 1) {
      const unsigned long long ok = __shfl_xor(key, off, 32);
      key = ok > key ? ok : key;
    }
    if (lane == 0) keys[((s & 1) << 5) + wid] = key;   // ping-pong buffer
    __syncthreads();

    // every wave redundantly reduces all 32 per-wave keys (kWaves == 32)
    key = keys[((s & 1) << 5) + lane];
#pragma unroll
    for (int off = 16; off > 0; off >>= 1) {
      const unsigned long long ok = __shfl_xor(key, off, 32);
      key = ok > key ? ok : key;
    }
    bi = key_index(key);

    // same-address LDS reads broadcast to all lanes
    lx = pts[3 * bi + 0];
    ly = pts[3 * bi + 1];
    lz = pts[3 * bi + 2];
    if (t == 0) {
      float* co = coords_out + ((size_t)b * kM + s) * 3;
      co[0] = lx; co[1] = ly; co[2] = lz;
      idx_out[b * kM + s] = bi;
    }
  }
}

// ---------------------------------------------------------------------------
// Kernel 2: feature gather. One wave per sample: 32 lanes x float4 = one
// 512 B row in b128 transactions, coalesced read of the gathered row and
// coalesced write. 8 samples per 256-thread block, 8192 blocks -> saturates
// HBM for this ~70 MB data-movement tail.
// ---------------------------------------------------------------------------
__global__ __launch_bounds__(256, 4) void gather_feats_kernel(
    const float* __restrict__ feats,   // [kN, kL, kD]
    const int* __restrict__ idx,       // [kN, kM]
    float* __restrict__ feats_out)     // [kN, kM, kD]
{
  const int wid  = threadIdx.x >> 5;
  const int lane = threadIdx.x & 31;
  const int s = blockIdx.x * 8 + wid;  // 0 .. kN*kM-1
  const int b = s / kM;
  const int j = idx[s];
  const float4* src =
      (const float4*)(feats + ((size_t)b * kL + j) * kD);
  float4* dst = (float4*)(feats_out + (size_t)s * kD);
  dst[lane] = src[lane];               // kD/4 == 32 == one float4 per lane
}

extern "C" void kernel_launch(void* const* d_in, const int* in_sizes, int n_in,
                              void* d_out, int out_size, void* d_ws,
                              size_t ws_size, hipStream_t stream) {
  (void)in_sizes; (void)n_in; (void)out_size; (void)ws_size;

  const float* coords = (const float*)d_in[0];   // [32, 8192, 3]
  const float* feats  = (const float*)d_in[1];   // [32, 8192, 128]

  float* out        = (float*)d_out;
  float* coords_out = out;                                   // [32, 2048, 3]
  float* feats_out  = out + (size_t)kN * kM * 3;             // [32, 2048, 128]
  int*   idx_ws     = (int*)d_ws;                            // [32, 2048]

  const size_t smem_bytes =
      (size_t)(3 * kL) * sizeof(float)          // staged cloud (98304 B)
      + 2 * kWaves * sizeof(unsigned long long);  // ping-pong key buffers

  fps_kernel<<<kN, kBlk, smem_bytes, stream>>>(coords, coords_out, idx_ws);
  gather_feats_kernel<<<(kN * kM) / 8, 256, 0, stream>>>(feats, idx_ws,
                                                         feats_out);
}